// DSBlock_13443247636681
// MI455X (gfx1250) — compile-verified
//
#include <hip/hip_runtime.h>
#include <hip/hip_bf16.h>

typedef float v8f __attribute__((ext_vector_type(8)));
typedef float v2f __attribute__((ext_vector_type(2)));

#define Bb 16
#define Cc 128
#define Nn 8192
#define Kk 256
#define KNN 6
#define Pp 128
#define IN_EPS 1e-3f
#define BN_EPS 1e-5f

__device__ __forceinline__ v8f wmma4(v2f a, v2f b, v8f c) {
  // D = A(16x4 f32) * B(4x16 f32) + C(16x16 f32)
  return __builtin_amdgcn_wmma_f32_16x16x4_f32(false, a, false, b, (short)0, c, false, false);
}
__device__ __forceinline__ v2f ld2(const float* p) { return *(const v2f*)p; }

// ---------------- per-(b,c) instance-norm stats ----------------
__global__ void k_instnorm_stats(const float* __restrict__ x, float* __restrict__ m, float* __restrict__ v) {
  int bc = blockIdx.x;
  const float* row = x + (size_t)bc * Nn;
  float s = 0.f, ss = 0.f;
  for (int i = threadIdx.x; i < Nn; i += 256) { float t = row[i]; s += t; ss += t * t; }
  __shared__ float sh[256], sh2[256];
  sh[threadIdx.x] = s; sh2[threadIdx.x] = ss; __syncthreads();
  for (int o = 128; o; o >>= 1) {
    if (threadIdx.x < o) { sh[threadIdx.x] += sh[threadIdx.x + o]; sh2[threadIdx.x] += sh2[threadIdx.x + o]; }
    __syncthreads();
  }
  if (threadIdx.x == 0) {
    float mean = sh[0] / (float)Nn;
    m[bc] = mean;
    v[bc] = sh2[0] / (float)Nn - mean * mean;
  }
}

// ---------------- fold IN+BN into per-(b,c) affine ----------------
__global__ void k_affine(const float* __restrict__ m, const float* __restrict__ v,
                         const float* __restrict__ dpg, const float* __restrict__ dpb,
                         const float* __restrict__ dug, const float* __restrict__ dub,
                         float* adp, float* ddp, float* adu, float* ddu) {
  int c = threadIdx.x;
  if (c >= Cc) return;
  float bnv = 0.f;
  for (int b = 0; b < Bb; ++b) { float vb = v[b * Cc + c]; bnv += vb / (vb + IN_EPS); }
  bnv *= (1.f / Bb);
  float s2 = rsqrtf(bnv + BN_EPS);
  for (int b = 0; b < Bb; ++b) {
    float vb = v[b * Cc + c], mb = m[b * Cc + c];
    float s1 = rsqrtf(vb + IN_EPS);
    float a1 = dpg[c] * s1 * s2; adp[b * Cc + c] = a1; ddp[b * Cc + c] = dpb[c] - a1 * mb;
    float a2 = dug[c] * s1 * s2; adu[b * Cc + c] = a2; ddu[b * Cc + c] = dub[c] - a2 * mb;
  }
}

// ---------------- embed GEMM: out[b][kk][n] = W @ relu(a*x+d) + bias ----------------
// A tile: 64x128 (stride 132, b64-aligned, conflict-free).
// B tile: paired layout, pair-row q holds (k=2q, k=2q+1) interleaved per column:
//         Bs[q*66 + n*2 + parity]  -> one aligned b64 per B-fragment.
__global__ __launch_bounds__(256) void k_embed(const float* __restrict__ x, const float* __restrict__ W,
                                               const float* __restrict__ bias,
                                               const float* __restrict__ av, const float* __restrict__ dv,
                                               float* __restrict__ out) {
  const int b = blockIdx.z;
  const int Mbase = blockIdx.y * 64;   // over Kk
  const int Nbase = blockIdx.x * 32;   // over Nn
  __shared__ float As[64 * 132];
  __shared__ float Bs[64 * 66];
  for (int i = threadIdx.x; i < 64 * 128; i += 256) {
    int mm = i >> 7, kk2 = i & 127;
    As[mm * 132 + kk2] = W[(Mbase + mm) * Cc + kk2];
  }
  const float* xb = x + (size_t)b * Cc * Nn;
  const float* ab = av + b * Cc;
  const float* db = dv + b * Cc;
  for (int i = threadIdx.x; i < 128 * 32; i += 256) {
    int cc2 = i >> 5, nn2 = i & 31;
    float t = fmaf(ab[cc2], xb[(size_t)cc2 * Nn + Nbase + nn2], db[cc2]);
    Bs[(cc2 >> 1) * 66 + nn2 * 2 + (cc2 & 1)] = fmaxf(t, 0.f);
  }
  __syncthreads();
  int wave = threadIdx.x >> 5, lane = threadIdx.x & 31;
  int half = lane >> 4, lm = lane & 15;
  int m0 = (wave & 3) * 16, n0 = (wave >> 2) * 16;
  v8f acc = {};
  for (int k0 = 0; k0 < 128; k0 += 4) {
    v2f af = ld2(&As[(m0 + lm) * 132 + k0 + 2 * half]);
    v2f bf = ld2(&Bs[((k0 >> 1) + half) * 66 + (n0 + lm) * 2]);
    acc = wmma4(af, bf, acc);
  }
  float* ob = out + (size_t)b * Kk * Nn;
  for (int r = 0; r < 8; ++r) {
    int row = Mbase + m0 + r + (half ? 8 : 0);
    ob[(size_t)row * Nn + Nbase + n0 + lm] = acc[r] + bias[row];
  }
}

// ---------------- softmax over N per (b,kk), in place ----------------
__global__ void k_softmax_rows(float* __restrict__ buf) {
  float* p = buf + (size_t)blockIdx.x * Nn;
  __shared__ float sh[256];
  float mx = -3.4e38f;
  for (int i = threadIdx.x; i < Nn; i += 256) mx = fmaxf(mx, p[i]);
  sh[threadIdx.x] = mx; __syncthreads();
  for (int o = 128; o; o >>= 1) { if (threadIdx.x < o) sh[threadIdx.x] = fmaxf(sh[threadIdx.x], sh[threadIdx.x + o]); __syncthreads(); }
  mx = sh[0]; __syncthreads();
  float s = 0.f;
  for (int i = threadIdx.x; i < Nn; i += 256) { float e = __expf(p[i] - mx); p[i] = e; s += e; }
  sh[threadIdx.x] = s; __syncthreads();
  for (int o = 128; o; o >>= 1) { if (threadIdx.x < o) sh[threadIdx.x] += sh[threadIdx.x + o]; __syncthreads(); }
  float inv = 1.f / sh[0];
  for (int i = threadIdx.x; i < Nn; i += 256) p[i] *= inv;
}

// ---------------- x_down[b][p][kk] = sum_n x[b][p][n] * S[b][kk][n] ----------------
__global__ __launch_bounds__(256) void k_xdown(const float* __restrict__ x, const float* __restrict__ S,
                                               float* __restrict__ xd) {
  int b = blockIdx.z;
  int Mbase = blockIdx.y * 64;   // p rows
  int Nbase = blockIdx.x * 32;   // kk cols
  __shared__ float As[64 * 36];
  __shared__ float Bs[16 * 66];
  const float* xb = x + (size_t)b * Cc * Nn;
  const float* Sb = S + (size_t)b * Kk * Nn;
  int wave = threadIdx.x >> 5, lane = threadIdx.x & 31;
  int half = lane >> 4, lm = lane & 15;
  int m0 = (wave & 3) * 16, n0 = (wave >> 2) * 16;
  v8f acc = {};
  for (int t0 = 0; t0 < Nn; t0 += 32) {
    __syncthreads();
    {
      int mm = threadIdx.x >> 5, kc = threadIdx.x & 31;
      if (t0 + 32 < Nn) {
        __builtin_prefetch(&xb[(size_t)(Mbase + mm) * Nn + t0 + 32 + kc], 0, 3);
        __builtin_prefetch(&Sb[(size_t)(Nbase + mm) * Nn + t0 + 32 + kc], 0, 3);
      }
    }
    for (int i = threadIdx.x; i < 64 * 32; i += 256) {
      int mm = i >> 5, kc = i & 31;
      As[mm * 36 + kc] = xb[(size_t)(Mbase + mm) * Nn + t0 + kc];
    }
    for (int i = threadIdx.x; i < 32 * 32; i += 256) {
      int col = i >> 5, kc = i & 31;
      Bs[(kc >> 1) * 66 + col * 2 + (kc & 1)] = Sb[(size_t)(Nbase + col) * Nn + t0 + kc];
    }
    __syncthreads();
    for (int k0 = 0; k0 < 32; k0 += 4) {
      v2f af = ld2(&As[(m0 + lm) * 36 + k0 + 2 * half]);
      v2f bf = ld2(&Bs[((k0 >> 1) + half) * 66 + (n0 + lm) * 2]);
      acc = wmma4(af, bf, acc);
    }
  }
  float* xdb = xd + (size_t)b * Cc * Kk;
  for (int r = 0; r < 8; ++r) {
    int row = Mbase + m0 + r + (half ? 8 : 0);
    xdb[row * Kk + Nbase + n0 + lm] = acc[r];
  }
}

// ---------------- KNN top-6 in 256-d feature space over 128 points ----------------
__global__ void k_knn(const float* __restrict__ xd, int* __restrict__ idx) {
  int b = blockIdx.x >> 7;
  int i = blockIdx.x & 127;
  int j = threadIdx.x;            // 128 threads
  const float* base = xd + (size_t)b * Cc * Kk;
  const float* ri = base + i * Kk;
  const float* rj = base + j * Kk;
  float dij = 0.f, djj = 0.f;
  for (int ch = 0; ch < Kk; ++ch) { float aj = rj[ch]; dij = fmaf(ri[ch], aj, dij); djj = fmaf(aj, aj, djj); }
  __shared__ float sq[128];
  __shared__ float pd[128];
  sq[j] = djj; __syncthreads();
  pd[j] = 2.f * dij - sq[i] - djj;
  __syncthreads();
  if (j == 0) {
    int chosen[KNN];
    for (int t = 0; t < KNN; ++t) {
      float best = -3.4e38f; int bi = 0;
      for (int q = 0; q < 128; ++q) {
        bool used = false;
        for (int u = 0; u < t; ++u) if (chosen[u] == q) used = true;
        if (!used && pd[q] > best) { best = pd[q]; bi = q; }
      }
      chosen[t] = bi;
      idx[(b * Pp + i) * KNN + t] = bi;
    }
  }
}

// ---------------- transpose / combine edge-conv weights ----------------
__global__ void k_prepw(const float* __restrict__ w1, const float* __restrict__ w2,
                        float* __restrict__ WcT, float* __restrict__ WnT, float* __restrict__ W2T) {
  int t = blockIdx.x * 256 + threadIdx.x;   // 65536
  int o = t >> 8, c = t & 255;
  float wa = w1[o * 512 + c], wb = w1[o * 512 + 256 + c];
  WcT[c * 256 + o] = wa + wb;
  WnT[c * 256 + o] = -wb;
  W2T[c * 256 + o] = w2[o * 256 + c];
}

// ---------------- edge conv1: out1[b][o][p][j] ----------------
__global__ __launch_bounds__(256) void k_conv1(const float* __restrict__ xd, const int* __restrict__ idx,
                                               const float* __restrict__ WcT, const float* __restrict__ WnT,
                                               const float* __restrict__ b1, float* __restrict__ out1) {
  int col = blockIdx.x;                       // b*Pp*KNN
  int j = col % KNN, p = (col / KNN) % Pp, b = col / (KNN * Pp);
  __shared__ float ctr[256], ngh[256];
  int nb = idx[col];
  const float* base = xd + (size_t)b * Cc * Kk;
  ctr[threadIdx.x] = base[p * Kk + threadIdx.x];
  ngh[threadIdx.x] = base[nb * Kk + threadIdx.x];
  __syncthreads();
  int o = threadIdx.x;
  float s = b1[o];
  for (int c = 0; c < 256; ++c)
    s = fmaf(WcT[c * 256 + o], ctr[c], fmaf(WnT[c * 256 + o], ngh[c], s));
  out1[(((size_t)b * Kk + o) * Pp + p) * KNN + j] = s;
}

// ---------------- training-mode BN stats per channel over (B,P,k) ----------------
__global__ void k_bnstats(const float* __restrict__ src, float* __restrict__ mu, float* __restrict__ inv) {
  int o = blockIdx.x;
  float s = 0.f, ss = 0.f;
  for (int b = 0; b < Bb; ++b) {
    const float* p = src + ((size_t)b * Kk + o) * Pp * KNN;
    for (int i = threadIdx.x; i < Pp * KNN; i += 256) { float t = p[i]; s += t; ss += t * t; }
  }
  __shared__ float sh[256], sh2[256];
  sh[threadIdx.x] = s; sh2[threadIdx.x] = ss; __syncthreads();
  for (int w = 128; w; w >>= 1) {
    if (threadIdx.x < w) { sh[threadIdx.x] += sh[threadIdx.x + w]; sh2[threadIdx.x] += sh2[threadIdx.x + w]; }
    __syncthreads();
  }
  if (threadIdx.x == 0) {
    const float n = (float)(Bb * Pp * KNN);
    float mean = sh[0] / n;
    mu[o] = mean;
    inv[o] = rsqrtf(sh2[0] / n - mean * mean + BN_EPS);
  }
}

// ---------------- edge conv2 with fused BN1+relu on input column ----------------
__global__ __launch_bounds__(256) void k_conv2(const float* __restrict__ out1,
                                               const float* __restrict__ mu1, const float* __restrict__ inv1,
                                               const float* __restrict__ g1, const float* __restrict__ bb1,
                                               const float* __restrict__ W2T, const float* __restrict__ b2,
                                               float* __restrict__ out2) {
  int col = blockIdx.x;
  int j = col % KNN, p = (col / KNN) % Pp, b = col / (KNN * Pp);
  __shared__ float u[256];
  int c = threadIdx.x;
  float t = out1[(((size_t)b * Kk + c) * Pp + p) * KNN + j];
  t = fmaf((t - mu1[c]) * inv1[c], g1[c], bb1[c]);
  u[c] = fmaxf(t, 0.f);
  __syncthreads();
  int o = c;
  float s = b2[o];
  for (int cc = 0; cc < 256; ++cc) s = fmaf(W2T[cc * 256 + o], u[cc], s);
  out2[(((size_t)b * Kk + o) * Pp + p) * KNN + j] = s;
}

// ---------------- BN2+relu then max over k neighbors: gmax[b][o][p] ----------------
__global__ void k_maxk(const float* __restrict__ out2,
                       const float* __restrict__ mu2, const float* __restrict__ inv2,
                       const float* __restrict__ g2, const float* __restrict__ bb2,
                       float* __restrict__ gmax) {
  int t = blockIdx.x * 256 + threadIdx.x;   // Bb*Kk*Pp
  int p = t % Pp, o = (t / Pp) % Kk, b = t / (Pp * Kk);
  const float* src = out2 + ((size_t)b * Kk + o) * Pp * KNN + (size_t)p * KNN;
  float a = inv2[o] * g2[o];
  float d = bb2[o] - mu2[o] * a;
  float mx = -3.4e38f;
  for (int j = 0; j < KNN; ++j) mx = fmaxf(mx, fmaxf(fmaf(src[j], a, d), 0.f));
  gmax[t] = mx;
}

// ---------------- T[b][oc][k] = sum_p sc_wB[oc][p] * gmax[b][k][p] ----------------
__global__ void k_tmat(const float* __restrict__ gmax, const float* __restrict__ scw, float* __restrict__ T) {
  int t = blockIdx.x * 256 + threadIdx.x;   // Bb*Cc*Kk
  int k = t % Kk, oc = (t / Kk) % Cc, b = t / (Kk * Cc);
  const float* g = gmax + ((size_t)b * Kk + k) * Pp;
  const float* w = scw + oc * 256 + 128;
  float s = 0.f;
  for (int p = 0; p < Pp; ++p) s = fmaf(w[p], g[p], s);
  T[((size_t)b * Cc + oc) * Kk + k] = s;
}

// ---------------- softmax over K per (b,n), in place ----------------
__global__ void k_softmax_cols(float* __restrict__ buf) {
  int t = blockIdx.x * 256 + threadIdx.x;   // Bb*Nn
  int n = t % Nn, b = t / Nn;
  float* p = buf + (size_t)b * Kk * Nn + n;
  float mx = -3.4e38f;
  for (int k = 0; k < Kk; ++k) mx = fmaxf(mx, p[(size_t)k * Nn]);
  float s = 0.f;
  for (int k = 0; k < Kk; ++k) { float e = __expf(p[(size_t)k * Nn] - mx); p[(size_t)k * Nn] = e; s += e; }
  float inv = 1.f / s;
  for (int k = 0; k < Kk; ++k) p[(size_t)k * Nn] *= inv;
}

// ---------------- final: out = sc_wA@x + T@S2 + sc_b  (two K segments) ----------------
__global__ __launch_bounds__(256) void k_final(const float* __restrict__ x, const float* __restrict__ S2,
                                               const float* __restrict__ T, const float* __restrict__ scw,
                                               const float* __restrict__ scb, float* __restrict__ out) {
  int b = blockIdx.z;
  int Mbase = blockIdx.y * 64;   // over Cc
  int Nbase = blockIdx.x * 32;   // over Nn
  __shared__ float As[64 * 36];
  __shared__ float Bs[16 * 66];
  const float* xb = x + (size_t)b * Cc * Nn;
  const float* Sb = S2 + (size_t)b * Kk * Nn;
  const float* Tb = T + (size_t)b * Cc * Kk;
  int wave = threadIdx.x >> 5, lane = threadIdx.x & 31;
  int half = lane >> 4, lm = lane & 15;
  int m0 = (wave & 3) * 16, n0 = (wave >> 2) * 16;
  v8f acc = {};
  for (int chunk = 0; chunk < 12; ++chunk) {
    __syncthreads();
    if (chunk < 4) {
      int c0 = chunk * 32;
      if (chunk + 1 < 4) {
        int kc = threadIdx.x >> 3, nn2 = (threadIdx.x & 7) * 4;
        __builtin_prefetch(&xb[(size_t)(c0 + 32 + kc) * Nn + Nbase + nn2], 0, 3);
      }
      for (int i = threadIdx.x; i < 64 * 32; i += 256) {
        int mm = i >> 5, kc = i & 31;
        As[mm * 36 + kc] = scw[(Mbase + mm) * 256 + c0 + kc];
      }
      for (int i = threadIdx.x; i < 32 * 32; i += 256) {
        int kc = i >> 5, nn2 = i & 31;
        Bs[(kc >> 1) * 66 + nn2 * 2 + (kc & 1)] = xb[(size_t)(c0 + kc) * Nn + Nbase + nn2];
      }
    } else {
      int k0g = (chunk - 4) * 32;
      if (chunk + 1 < 12) {
        int kc = threadIdx.x >> 3, nn2 = (threadIdx.x & 7) * 4;
        int kn = (chunk + 1 < 12) ? ((chunk - 3) * 32) : 0;
        __builtin_prefetch(&Sb[(size_t)(kn + kc) * Nn + Nbase + nn2], 0, 3);
      }
      for (int i = threadIdx.x; i < 64 * 32; i += 256) {
        int mm = i >> 5, kc = i & 31;
        As[mm * 36 + kc] = Tb[(Mbase + mm) * Kk + k0g + kc];
      }
      for (int i = threadIdx.x; i < 32 * 32; i += 256) {
        int kc = i >> 5, nn2 = i & 31;
        Bs[(kc >> 1) * 66 + nn2 * 2 + (kc & 1)] = Sb[(size_t)(k0g + kc) * Nn + Nbase + nn2];
      }
    }
    __syncthreads();
    for (int k0 = 0; k0 < 32; k0 += 4) {
      v2f af = ld2(&As[(m0 + lm) * 36 + k0 + 2 * half]);
      v2f bf = ld2(&Bs[((k0 >> 1) + half) * 66 + (n0 + lm) * 2]);
      acc = wmma4(af, bf, acc);
    }
  }
  float* ob = out + (size_t)b * Cc * Nn;
  for (int r = 0; r < 8; ++r) {
    int row = Mbase + m0 + r + (half ? 8 : 0);
    ob[(size_t)row * Nn + Nbase + n0 + lm] = acc[r] + scb[row];
  }
}

extern "C" void kernel_launch(void* const* d_in, const int* in_sizes, int n_in,
                              void* d_out, int out_size, void* d_ws, size_t ws_size,
                              hipStream_t stream) {
  const float* x        = (const float*)d_in[0];
  const float* dp_gamma = (const float*)d_in[1];
  const float* dp_beta  = (const float*)d_in[2];
  const float* dp_w     = (const float*)d_in[3];
  const float* dp_b     = (const float*)d_in[4];
  const float* du_gamma = (const float*)d_in[5];
  const float* du_beta  = (const float*)d_in[6];
  const float* du_w     = (const float*)d_in[7];
  const float* du_b     = (const float*)d_in[8];
  const float* dg_w1    = (const float*)d_in[9];
  const float* dg_b1    = (const float*)d_in[10];
  const float* dg_g1    = (const float*)d_in[11];
  const float* dg_bb1   = (const float*)d_in[12];
  const float* dg_w2    = (const float*)d_in[13];
  const float* dg_b2    = (const float*)d_in[14];
  const float* dg_g2    = (const float*)d_in[15];
  const float* dg_bb2   = (const float*)d_in[16];
  const float* sc_w     = (const float*)d_in[17];
  const float* sc_b     = (const float*)d_in[18];
  float* out = (float*)d_out;

  float* ws = (float*)d_ws;
  size_t off = 0;
  float* buf1 = ws + off; off += (size_t)Bb * Kk * Nn;     // embed / S / embed2 / S2
  float* m    = ws + off; off += Bb * Cc;
  float* v    = ws + off; off += Bb * Cc;
  float* adp  = ws + off; off += Bb * Cc;
  float* ddp  = ws + off; off += Bb * Cc;
  float* adu  = ws + off; off += Bb * Cc;
  float* ddu  = ws + off; off += Bb * Cc;
  float* xd   = ws + off; off += (size_t)Bb * Cc * Kk;
  int*   idx  = (int*)(ws + off); off += Bb * Pp * KNN;
  float* WcT  = ws + off; off += 256 * 256;
  float* WnT  = ws + off; off += 256 * 256;
  float* W2T  = ws + off; off += 256 * 256;
  float* out1 = ws + off; off += (size_t)Bb * Kk * Pp * KNN;
  float* out2 = ws + off; off += (size_t)Bb * Kk * Pp * KNN;
  float* mu1  = ws + off; off += Kk;
  float* inv1 = ws + off; off += Kk;
  float* mu2  = ws + off; off += Kk;
  float* inv2 = ws + off; off += Kk;
  float* gmax = ws + off; off += (size_t)Bb * Kk * Pp;
  float* Tm   = ws + off; off += (size_t)Bb * Cc * Kk;

  // 1) instance-norm stats + folded affines + transposed edge-conv weights
  k_instnorm_stats<<<Bb * Cc, 256, 0, stream>>>(x, m, v);
  k_affine<<<1, 128, 0, stream>>>(m, v, dp_gamma, dp_beta, du_gamma, du_beta, adp, ddp, adu, ddu);
  k_prepw<<<256, 256, 0, stream>>>(dg_w1, dg_w2, WcT, WnT, W2T);

  // 2) diff_pool: embed (WMMA), softmax over N, x_down (WMMA)
  k_embed<<<dim3(Nn / 32, Kk / 64, Bb), 256, 0, stream>>>(x, dp_w, dp_b, adp, ddp, buf1);
  k_softmax_rows<<<Bb * Kk, 256, 0, stream>>>(buf1);
  k_xdown<<<dim3(Kk / 32, Cc / 64, Bb), 256, 0, stream>>>(x, buf1, xd);

  // 3) DGCNN block: knn, edge conv1+BN, conv2+BN, max over k
  k_knn<<<Bb * Pp, 128, 0, stream>>>(xd, idx);
  k_conv1<<<Bb * Pp * KNN, 256, 0, stream>>>(xd, idx, WcT, WnT, dg_b1, out1);
  k_bnstats<<<Kk, 256, 0, stream>>>(out1, mu1, inv1);
  k_conv2<<<Bb * Pp * KNN, 256, 0, stream>>>(out1, mu1, inv1, dg_g1, dg_bb1, W2T, dg_b2, out2);
  k_bnstats<<<Kk, 256, 0, stream>>>(out2, mu2, inv2);
  k_maxk<<<(Bb * Kk * Pp) / 256, 256, 0, stream>>>(out2, mu2, inv2, dg_g2, dg_bb2, gmax);
  k_tmat<<<(Bb * Cc * Kk) / 256, 256, 0, stream>>>(gmax, sc_w, Tm);

  // 4) diff_unpool: embed2 (WMMA, reuse buf1), softmax over K, fused final GEMM
  k_embed<<<dim3(Nn / 32, Kk / 64, Bb), 256, 0, stream>>>(x, du_w, du_b, adu, ddu, buf1);
  k_softmax_cols<<<(Bb * Nn) / 256, 256, 0, stream>>>(buf1);
  k_final<<<dim3(Nn / 32, Cc / 64, Bb), 256, 0, stream>>>(x, buf1, Tm, sc_w, sc_b, out);
}